// RetinaNetRT_790273982881
// MI455X (gfx1250) — compile-verified
//
#include <hip/hip_runtime.h>
#include <math.h>

#define NUM_LEVELS 5
#define BATCH      8
#define NCLS       80
#define ROWSTRIDE  84          // 4 reg + 80 cls floats per anchor
#define NBUCKET    8192        // histogram buckets (float bits >> 18)
#define NCAND      4441        // 1000*4 + 441 candidates per image
#define NSORT      8192        // padded pow2 for bitonic sort
#define COLBLOCKS  139         // ceil(4441/32)
#define ROWTILES   278         // ceil(4441/16)
#define MAXDET     100
#define SCORE_T    0.05f
#define NMS_T      0.5f
#define SCALE_CLAMP_F 4.135166556742356f   // log(1000/16)

#define CHUNK_A    48                      // anchors per TDM chunk
#define CHUNKF     (CHUNK_A*ROWSTRIDE)     // 4032 floats = 16128 B
#define HIST_SMEM  (NBUCKET*4 + 2*CHUNKF*4)  // 32768 + 32256 = 65024 B

typedef __attribute__((ext_vector_type(2))) float v2f;
typedef __attribute__((ext_vector_type(8))) float v8f;
typedef __attribute__((ext_vector_type(4))) unsigned int u32x4;
typedef __attribute__((ext_vector_type(4))) int i32x4;
typedef __attribute__((ext_vector_type(8))) int i32x8;

// ---- Tensor Data Mover availability (probe-verified 5-arg form on ROCm 7.2) ----
#if __has_builtin(__builtin_amdgcn_tensor_load_to_lds) && __has_builtin(__builtin_amdgcn_s_wait_tensorcnt)
  #define HAVE_TDM 1
  #if __has_include(<hip/amd_detail/amd_gfx1250_TDM.h>)
    #define TDM_SIX_ARGS 1
  #else
    #define TDM_SIX_ARGS 0
  #endif
#else
  #define HAVE_TDM 0
#endif

struct KArgs { const float* out[NUM_LEVELS]; const float* anc[NUM_LEVELS]; };

__device__ __forceinline__ int lvA(int l){
  switch(l){case 0:return 90000;case 1:return 22500;case 2:return 5625;case 3:return 1521;default:return 441;}
}
__device__ __forceinline__ int lvK(int l){ return (l==4)?441:1000; }
__device__ __forceinline__ int lvOff(int l){ return l*1000; }

__device__ __forceinline__ int bucket_of(unsigned u){
  if (u & 0x80000000u) return 0;           // negative scores -> lowest bucket
  unsigned b = u >> 18;                    // sign+exp+5 mantissa bits
  return (int)(b < (unsigned)(NBUCKET-1) ? b : (unsigned)(NBUCKET-1));
}

#if HAVE_TDM
// Issue a 1-D TDM copy of nf floats (contiguous) from gsrc into LDS byte offset ldsOff.
// D# group layout per CDNA5 ISA 8.3-8.6: group0 = {count/type/addr}, group1 = dims/tile.
__device__ __forceinline__ void tdm_issue_1d(const float* gsrc, unsigned nf, unsigned ldsOff) {
  const unsigned long long ga = (unsigned long long)(uintptr_t)gsrc;
  u32x4 g0;
  g0.x = 1u;                                               // count=1, is_restore=0, gather off
  g0.y = ldsOff;                                           // lds_addr (bytes)
  g0.z = (unsigned)(ga & 0xFFFFFFFFull);                   // global_addr[31:0]
  g0.w = (unsigned)((ga >> 32) & 0x01FFFFFFull) | 0x80000000u;  // global_addr[56:32] | type=2
  i32x8 g1;
  g1[0] = 0x20000;                                         // wg_mask=0 (no cluster), data_size=4B
  g1[1] = (int)((nf & 0xFFFFu) << 16);                     // tensor_dim0[15:0]
  g1[2] = (int)((nf >> 16) | (1u << 16));                  // tensor_dim0[31:16] | tensor_dim1=1 (lo)
  g1[3] = (int)(nf << 16);                                 // tensor_dim1 hi=0 | tile_dim0=nf
  g1[4] = 1;                                               // tile_dim1=1, tile_dim2=0 (1-D tile)
  g1[5] = (int)nf;                                         // tensor_dim0_stride[31:0]
  g1[6] = (int)((nf & 0xFFFFu) << 16);                     // stride0 hi=0 | stride1[15:0]=nf
  g1[7] = 0;                                               // stride1[47:16]
  i32x4 g2 = {0,0,0,0};                                    // dims 2/3 unused
  i32x4 g3 = {0,0,0,0};
#if TDM_SIX_ARGS
  i32x8 g4 = {0,0,0,0,0,0,0,0};
  __builtin_amdgcn_tensor_load_to_lds(g0, g1, g2, g3, g4, 0);
#else
  __builtin_amdgcn_tensor_load_to_lds(g0, g1, g2, g3, 0);
#endif
}
#endif

// ---------------------------------------------------------------- init
__global__ void init_kernel(unsigned* hist, int* sel, float* cscore) {
  int tid = blockIdx.x*blockDim.x + threadIdx.x;
  int nth = gridDim.x*blockDim.x;
  for (int i = tid; i < BATCH*NUM_LEVELS*NBUCKET; i += nth) hist[i] = 0u;
  for (int i = tid; i < BATCH*NUM_LEVELS*4;       i += nth) sel[i]  = 0;
  for (int i = tid; i < BATCH*NSORT;              i += nth) cscore[i] = -3.0e38f; // sort padding
}

// ---------------------------------------------------------------- pass 1: score histogram
// TDM double-buffers anchor rows into LDS while VALUs bucket the previous chunk.
__global__ void hist_kernel(KArgs args, unsigned* __restrict__ hist) {
  extern __shared__ unsigned smem[];
  unsigned* lhist = smem;                       // NBUCKET words
  const int lvl = blockIdx.y, img = blockIdx.z;
  const int A = lvA(lvl);
  for (int i = threadIdx.x; i < NBUCKET; i += blockDim.x) lhist[i] = 0u;
  __syncthreads();

  const float* base = args.out[lvl] + (size_t)img * A * ROWSTRIDE;
  const int span = (A + gridDim.x - 1) / gridDim.x;
  const int a0 = min(A, (int)blockIdx.x * span);
  const int a1 = min(A, a0 + span);

#if HAVE_TDM
  float* stage = (float*)(smem + NBUCKET);      // 2 * CHUNKF floats
  const int nchunk = (a1 - a0 + CHUNK_A - 1) / CHUNK_A;
  const bool issuer = (threadIdx.x < 32);       // wave 0 owns the DMA (TDM ignores EXEC)
  if (nchunk > 0 && issuer) {
    int na = min(CHUNK_A, a1 - a0);
    tdm_issue_1d(base + (size_t)a0 * ROWSTRIDE, (unsigned)(na * ROWSTRIDE),
                 (unsigned)(uintptr_t)&stage[0]);
  }
  for (int ck = 0; ck < nchunk; ++ck) {
    const int ca = a0 + ck * CHUNK_A;
    const int na = min(CHUNK_A, a1 - ca);
    if (issuer) {
      if (ck + 1 < nchunk) {
        const int cb = a0 + (ck + 1) * CHUNK_A;
        const int nb = min(CHUNK_A, a1 - cb);
        tdm_issue_1d(base + (size_t)cb * ROWSTRIDE, (unsigned)(nb * ROWSTRIDE),
                     (unsigned)(uintptr_t)&stage[((ck + 1) & 1) * CHUNKF]);
        __builtin_amdgcn_s_wait_tensorcnt(1);   // chunk ck complete (in-order)
      } else {
        __builtin_amdgcn_s_wait_tensorcnt(0);
      }
    }
    __syncthreads();                            // publish staged chunk to all waves
    const float* cur = stage + (ck & 1) * CHUNKF;
    const int nscore = na * NCLS;
    for (int t = threadIdx.x; t < nscore; t += blockDim.x) {
      const int aa = t / NCLS, c = t - aa * NCLS;
      const float s = cur[aa * ROWSTRIDE + 4 + c];
      atomicAdd(&lhist[bucket_of(__float_as_uint(s))], 1u);
    }
    __syncthreads();                            // chunk consumed; buffer reusable
  }
#else
  for (int a = a0 + threadIdx.x; a < a1; a += blockDim.x) {
    const float4* sp = (const float4*)(base + (size_t)a*ROWSTRIDE + 4);
    __builtin_prefetch(base + (size_t)(a + blockDim.x)*ROWSTRIDE + 4, 0, 0);
    #pragma unroll 5
    for (int q = 0; q < 20; ++q) {
      float4 v = sp[q];
      atomicAdd(&lhist[bucket_of(__float_as_uint(v.x))], 1u);
      atomicAdd(&lhist[bucket_of(__float_as_uint(v.y))], 1u);
      atomicAdd(&lhist[bucket_of(__float_as_uint(v.z))], 1u);
      atomicAdd(&lhist[bucket_of(__float_as_uint(v.w))], 1u);
    }
  }
#endif
  __syncthreads();
  unsigned* gh = hist + (size_t)(img*NUM_LEVELS + lvl)*NBUCKET;
  for (int i = threadIdx.x; i < NBUCKET; i += blockDim.x)
    if (lhist[i]) atomicAdd(&gh[i], lhist[i]);
}

// ---------------------------------------------------------------- find per-(img,level) threshold bucket
__global__ void scan_kernel(const unsigned* __restrict__ hist, int* __restrict__ sel) {
  const int pair = blockIdx.x;                 // img*5 + lvl
  if (threadIdx.x != 0) return;
  const int K = lvK(pair % NUM_LEVELS);
  const unsigned* h = hist + (size_t)pair*NBUCKET;
  long long cum = 0; int t = 0; long long above = 0;
  for (int b = NBUCKET-1; b >= 0; --b) {
    cum += h[b];
    if (cum >= K) { t = b; above = cum - h[b]; break; }
  }
  int* s = sel + pair*4;
  s[0] = t; s[1] = (int)above; s[2] = 0; s[3] = 0;
}

// ---------------------------------------------------------------- pass 2: compact top-K + decode boxes
__global__ void select_kernel(KArgs args, int* __restrict__ sel,
                              float* __restrict__ cscore, float* __restrict__ cbox,
                              float* __restrict__ ccls) {
  const int lvl = blockIdx.y, img = blockIdx.z;
  const int A = lvA(lvl), K = lvK(lvl), off = lvOff(lvl);
  const int pair = img*NUM_LEVELS + lvl;
  const int thr   = sel[pair*4 + 0];
  const int above = sel[pair*4 + 1];
  const int need  = K - above;
  const float* base = args.out[lvl] + (size_t)img * A * ROWSTRIDE;
  const float* anc  = args.anc[lvl];

  const int stride = blockDim.x * gridDim.x;
  for (int a = blockIdx.x*blockDim.x + threadIdx.x; a < A; a += stride) {
    const float4* sp = (const float4*)(base + (size_t)a*ROWSTRIDE + 4);
    __builtin_prefetch(base + (size_t)(a + stride)*ROWSTRIDE + 4, 0, 0);
    #pragma unroll 5
    for (int q = 0; q < 20; ++q) {
      float4 v = sp[q];
      float vs[4] = {v.x, v.y, v.z, v.w};
      #pragma unroll
      for (int e = 0; e < 4; ++e) {
        const float s = vs[e];
        const int bkt = bucket_of(__float_as_uint(s));
        int slot = -1;
        if (bkt > thr) {
          slot = atomicAdd(&sel[pair*4 + 2], 1);
        } else if (bkt == thr) {
          int p = atomicAdd(&sel[pair*4 + 3], 1);
          if (p < need) slot = above + p;
        }
        if (slot >= 0) {
          const int c = q*4 + e;
          float4 reg = *(const float4*)(base + (size_t)a*ROWSTRIDE);    // dx dy dw dh
          float ax0 = anc[a*4+0], ay0 = anc[a*4+1], ax1 = anc[a*4+2], ay1 = anc[a*4+3];
          float wa = ax1-ax0, ha = ay1-ay0;
          float cxa = ax0 + 0.5f*wa, cya = ay0 + 0.5f*ha;
          float dw = fminf(reg.z, SCALE_CLAMP_F), dh = fminf(reg.w, SCALE_CLAMP_F);
          float pcx = reg.x*wa + cxa, pcy = reg.y*ha + cya;
          float pw = expf(dw)*wa, ph = expf(dh)*ha;
          const int g = img*NSORT + off + slot;
          cscore[g] = (s > SCORE_T) ? s : -1.0f;
          ccls[g]   = (float)c;
          cbox[(size_t)g*4+0] = pcx - 0.5f*pw;
          cbox[(size_t)g*4+1] = pcy - 0.5f*ph;
          cbox[(size_t)g*4+2] = pcx + 0.5f*pw;
          cbox[(size_t)g*4+3] = pcy + 0.5f*ph;
        }
      }
    }
  }
}

// ---------------------------------------------------------------- per-image bitonic sort (desc, stable) + max_coord
__global__ void __launch_bounds__(1024) sort_kernel(const float* __restrict__ cscore,
                                                    const float* __restrict__ cbox,
                                                    int* __restrict__ order,
                                                    float* __restrict__ maxcoord) {
  extern __shared__ unsigned long long keys[];   // NSORT * 8B = 64KB
  const int img = blockIdx.x;
  const int tid = threadIdx.x, nt = blockDim.x;

  for (int i = tid; i < NSORT; i += nt) {
    unsigned u = __float_as_uint(cscore[img*NSORT + i]);
    u = (u & 0x80000000u) ? ~u : (u | 0x80000000u);           // monotone key
    keys[i] = ((unsigned long long)u << 32) | (unsigned)(0xFFFFFFFFu - (unsigned)i); // stable desc
  }
  __syncthreads();

  for (int k = 2; k <= NSORT; k <<= 1) {
    for (int j = k >> 1; j > 0; j >>= 1) {
      for (int i = tid; i < NSORT; i += nt) {
        int ixj = i ^ j;
        if (ixj > i) {
          unsigned long long a = keys[i], b = keys[ixj];
          bool desc = ((i & k) == 0);
          if (desc ? (a < b) : (a > b)) { keys[i] = b; keys[ixj] = a; }
        }
      }
      __syncthreads();
    }
  }
  for (int i = tid; i < NSORT; i += nt)
    order[img*NSORT + i] = (int)(0xFFFFFFFFu - (unsigned)keys[i]);
  __syncthreads();

  // max over valid boxes (reference: max(where(valid, b, 0)) + 1)
  float m = 0.0f;
  for (int s = tid; s < NCAND; s += nt) {
    if (cscore[img*NSORT + s] > 0.0f) {
      const float* bp = cbox + (size_t)(img*NSORT + s)*4;
      m = fmaxf(m, fmaxf(fmaxf(bp[0], bp[1]), fmaxf(bp[2], bp[3])));
    }
  }
  float* red = (float*)keys;
  __syncthreads();
  red[tid] = m; __syncthreads();
  for (int s = nt >> 1; s > 0; s >>= 1) {
    if (tid < s) red[tid] = fmaxf(red[tid], red[tid + s]);
    __syncthreads();
  }
  if (tid == 0) maxcoord[img] = red[0] + 1.0f;
}

// ---------------------------------------------------------------- IoU suppression mask: WMMA area-sum outer product
__global__ void __launch_bounds__(32) mask_kernel(const float* __restrict__ cbox,
                                                  const float* __restrict__ ccls,
                                                  const int* __restrict__ order,
                                                  const float* __restrict__ maxcoord,
                                                  unsigned* __restrict__ mask) {
  __shared__ __align__(16) float cbx[32][4];
  __shared__ __align__(16) float rbx[16][4];
  __shared__ float car[32];
  __shared__ float rar[16];

  const int cbk = blockIdx.x, rt = blockIdx.y, img = blockIdx.z;
  const int lane = threadIdx.x;
  const int rowBase = rt * 16, colBase = cbk * 32;
  const float mc = maxcoord[img];
  const int* ord = order + img*NSORT;

  { // 32 column boxes (class-offset)
    int j = colBase + lane;
    float b0=0.f,b1=0.f,b2=0.f,b3=0.f;
    if (j < NCAND) {
      int sl = ord[j];
      const float* bp = cbox + (size_t)(img*NSORT + sl)*4;
      float off = ccls[img*NSORT + sl] * mc;
      b0 = bp[0]+off; b1 = bp[1]+off; b2 = bp[2]+off; b3 = bp[3]+off;
    }
    cbx[lane][0]=b0; cbx[lane][1]=b1; cbx[lane][2]=b2; cbx[lane][3]=b3;
    car[lane] = (b2-b0)*(b3-b1);
  }
  if (lane < 16) { // 16 row boxes
    int i = rowBase + lane;
    float b0=0.f,b1=0.f,b2=0.f,b3=0.f;
    if (i < NCAND) {
      int sl = ord[i];
      const float* bp = cbox + (size_t)(img*NSORT + sl)*4;
      float off = ccls[img*NSORT + sl] * mc;
      b0 = bp[0]+off; b1 = bp[1]+off; b2 = bp[2]+off; b3 = bp[3]+off;
    }
    rbx[lane][0]=b0; rbx[lane][1]=b1; rbx[lane][2]=b2; rbx[lane][3]=b3;
    rar[lane] = (b2-b0)*(b3-b1);
  }
  __syncthreads();

  // Rank-2 outer product on the matrix core: S[i][j] = area_i*1 + 1*area_j
  const int nl = lane & 15;
  v2f aop; aop.x = (lane < 16) ? rar[nl]       : 0.0f; aop.y = (lane < 16) ? 1.0f : 0.0f;
  v2f b0v; b0v.x = (lane < 16) ? 1.0f : 0.0f;  b0v.y = (lane < 16) ? car[nl]      : 0.0f;
  v2f b1v; b1v.x = (lane < 16) ? 1.0f : 0.0f;  b1v.y = (lane < 16) ? car[nl + 16] : 0.0f;
  v8f z = {0.f,0.f,0.f,0.f,0.f,0.f,0.f,0.f};
  v8f S0 = __builtin_amdgcn_wmma_f32_16x16x4_f32(false, aop, false, b0v, (short)0, z, false, false);
  v8f S1 = __builtin_amdgcn_wmma_f32_16x16x4_f32(false, aop, false, b1v, (short)0, z, false, false);

  const int mAdd = (lane < 16) ? 0 : 8;
  const float cx0 = cbx[nl][0],    cy0 = cbx[nl][1],    cx1 = cbx[nl][2],    cy1 = cbx[nl][3];
  const float dx0 = cbx[nl+16][0], dy0 = cbx[nl+16][1], dx1 = cbx[nl+16][2], dy1 = cbx[nl+16][3];

  #pragma unroll
  for (int r = 0; r < 8; ++r) {
    const int m  = r + mAdd;
    const int gi = rowBase + m;
    const float rx0 = rbx[m][0], ry0 = rbx[m][1], rx1 = rbx[m][2], ry1 = rbx[m][3];

    // tile 0: cols colBase + [0,16)
    float iw = fminf(rx1, cx1) - fmaxf(rx0, cx0);
    float ih = fminf(ry1, cy1) - fmaxf(ry0, cy0);
    float inter = fmaxf(iw, 0.f) * fmaxf(ih, 0.f);
    bool sup0 = (inter > NMS_T * (S0[r] - inter)) && ((colBase + nl) > gi);
    unsigned bal0 = __builtin_amdgcn_ballot_w32(sup0);

    // tile 1: cols colBase + [16,32)
    iw = fminf(rx1, dx1) - fmaxf(rx0, dx0);
    ih = fminf(ry1, dy1) - fmaxf(ry0, dy0);
    inter = fmaxf(iw, 0.f) * fmaxf(ih, 0.f);
    bool sup1 = (inter > NMS_T * (S1[r] - inter)) && ((colBase + 16 + nl) > gi);
    unsigned bal1 = __builtin_amdgcn_ballot_w32(sup1);

    unsigned word_lo = (bal0 & 0xFFFFu) | ((bal1 & 0xFFFFu) << 16);   // row rowBase+r
    unsigned word_hi = (bal0 >> 16)     | ((bal1 >> 16) << 16);       // row rowBase+8+r
    if (lane == 0) {
      const size_t rb = (size_t)img * NCAND;
      const int g0 = rowBase + r, g1 = rowBase + 8 + r;
      if (g0 < NCAND) mask[(rb + g0) * COLBLOCKS + cbk] = word_lo;
      if (g1 < NCAND) mask[(rb + g1) * COLBLOCKS + cbk] = word_hi;
    }
  }
}

// ---------------------------------------------------------------- greedy reduce (1 wave / image) + emit top-100
__global__ void __launch_bounds__(32) nms_reduce_kernel(const float* __restrict__ cscore,
                                                        const float* __restrict__ cbox,
                                                        const float* __restrict__ ccls,
                                                        const int* __restrict__ order,
                                                        const unsigned* __restrict__ mask,
                                                        float* __restrict__ out) {
  __shared__ float    sscore[NCAND];
  __shared__ unsigned remv[COLBLOCKS];
  __shared__ unsigned keptbits[COLBLOCKS];
  __shared__ int      klist[MAXDET];
  __shared__ int      plist[MAXDET];

  const int img = blockIdx.x;
  const int lane = threadIdx.x;
  const int* ord = order + img*NSORT;

  for (int w = lane; w < COLBLOCKS; w += 32) { remv[w] = 0u; keptbits[w] = 0u; }
  for (int i = lane; i < NCAND; i += 32) sscore[i] = cscore[img*NSORT + ord[i]];
  __syncthreads();

  int nkeep = 0;
  for (int i = 0; i < NCAND; ++i) {
    const float s = sscore[i];
    const unsigned rw = remv[i >> 5];
    const bool kept = (s > 0.0f) && !((rw >> (i & 31)) & 1u);
    if (kept) {
      if (lane == 0) { klist[nkeep] = i; keptbits[i >> 5] |= 1u << (i & 31); }
      ++nkeep;
      if (nkeep >= MAXDET) break;
      const unsigned* mrow = mask + ((size_t)img * NCAND + i) * COLBLOCKS;
      for (int w = lane; w < COLBLOCKS; w += 32) remv[w] |= mrow[w];
      __syncthreads();
    }
  }
  __syncthreads();

  if (lane == 0) {
    const int need = MAXDET - nkeep;
    int np = 0;
    for (int i = 0; i < NCAND && np < need; ++i)
      if (!((keptbits[i >> 5] >> (i & 31)) & 1u)) plist[np++] = i;
  }
  __syncthreads();

  // d_out: boxes [8][100][4] | scores [8][100] | classes [8][100]
  for (int k = lane; k < MAXDET; k += 32) {
    int i; float sc;
    if (k < nkeep) { i = klist[k]; sc = sscore[i]; }
    else           { i = plist[k - nkeep]; sc = -1.0f; }
    const int sl = ord[i];
    const float* bp = cbox + (size_t)(img*NSORT + sl)*4;
    const float cls = ccls[img*NSORT + sl];
    float* ob = out + ((size_t)img*MAXDET + k)*4;
    ob[0] = bp[0]; ob[1] = bp[1]; ob[2] = bp[2]; ob[3] = bp[3];
    out[BATCH*MAXDET*4 + img*MAXDET + k] = sc;
    out[BATCH*MAXDET*5 + img*MAXDET + k] = cls;
  }
}

// ---------------------------------------------------------------- launch
extern "C" void kernel_launch(void* const* d_in, const int* in_sizes, int n_in,
                              void* d_out, int out_size, void* d_ws, size_t ws_size,
                              hipStream_t stream) {
  (void)in_sizes; (void)n_in; (void)out_size; (void)ws_size;

  KArgs ka;
  for (int l = 0; l < NUM_LEVELS; ++l) {
    ka.out[l] = (const float*)d_in[l];
    ka.anc[l] = (const float*)d_in[NUM_LEVELS + l];
  }

  char* ws = (char*)d_ws;
  size_t o = 0;
  auto take = [&](size_t bytes) { size_t r = o; o += (bytes + 255) & ~(size_t)255; return r; };
  unsigned* hist   = (unsigned*)(ws + take((size_t)BATCH*NUM_LEVELS*NBUCKET*4));
  int*      sel    = (int*)     (ws + take((size_t)BATCH*NUM_LEVELS*4*4));
  float*    cscore = (float*)   (ws + take((size_t)BATCH*NSORT*4));
  float*    cbox   = (float*)   (ws + take((size_t)BATCH*NSORT*16));
  float*    ccls   = (float*)   (ws + take((size_t)BATCH*NSORT*4));
  int*      order  = (int*)     (ws + take((size_t)BATCH*NSORT*4));
  float*    maxc   = (float*)   (ws + take((size_t)BATCH*4));
  unsigned* mask   = (unsigned*)(ws + take((size_t)BATCH*NCAND*COLBLOCKS*4));
  float*    out    = (float*)d_out;

  init_kernel<<<1024, 256, 0, stream>>>(hist, sel, cscore);
  hist_kernel<<<dim3(32, NUM_LEVELS, BATCH), 256, HIST_SMEM, stream>>>(ka, hist);
  scan_kernel<<<BATCH*NUM_LEVELS, 32, 0, stream>>>(hist, sel);
  select_kernel<<<dim3(96, NUM_LEVELS, BATCH), 256, 0, stream>>>(ka, sel, cscore, cbox, ccls);
  sort_kernel<<<BATCH, 1024, NSORT*8, stream>>>(cscore, cbox, order, maxc);
  mask_kernel<<<dim3(COLBLOCKS, ROWTILES, BATCH), 32, 0, stream>>>(cbox, ccls, order, maxc, mask);
  nms_reduce_kernel<<<BATCH, 32, 0, stream>>>(cscore, cbox, ccls, order, mask, out);
}